// DependencyGNN_GAT_7421703487715
// MI455X (gfx1250) — compile-verified
//
#include <hip/hip_runtime.h>
#include <hip/hip_bf16.h>

// ---------------------------------------------------------------------------
// DependencyGNN_GAT for MI455X (gfx1250, wave32, WMMA).
// All GEMMs use V_WMMA_F32_16X16X4_F32 (f32 tensor path, no precision loss).
// h-features (hq, ax) and Fw are kept in K-transposed layouts so every
// B-operand pair in the hot attention GEMM is a single aligned b64 load.
// ---------------------------------------------------------------------------

#define BB   16
#define LL   512
#define HH   256
#define NHH  8
#define HDD  32
#define H3   768
#define BLR  8192          // B*L
#define NEGV (-9.0e15f)

typedef __attribute__((ext_vector_type(2))) float v2f;
typedef __attribute__((ext_vector_type(8))) float v8f;

__device__ __forceinline__ v8f wmma_f32(v2f a, v2f b, v8f c) {
  // D = A(16x4 f32) * B(4x16 f32) + C(16x16 f32)
  return __builtin_amdgcn_wmma_f32_16x16x4_f32(
      false, a, false, b, (short)0, c, false, false);
}

// ------------------------------ small kernels ------------------------------

__global__ void relay_mean_kernel(const float* __restrict__ data,
                                  float* __restrict__ relay) {
  int b = blockIdx.x, h = threadIdx.x;                 // block 256
  const float* p = data + ((size_t)b * LL) * HH + h;
  float s = 0.f;
  for (int l = 0; l < LL; ++l) s += p[(size_t)l * HH];
  relay[b * HH + h] = s * (1.0f / (float)LL);
}

// axT[b][feat][l]: feat=2k -> data[b][l][k], feat=2k+1 -> relay[b][k]
__global__ void build_axT_kernel(const float* __restrict__ data,
                                 const float* __restrict__ relay,
                                 float* __restrict__ axT) {
  int b = blockIdx.x >> 9, feat = blockIdx.x & 511;    // grid B*512, block 256
  int k = feat >> 1;
  bool isrelay = (feat & 1) != 0;
  float rv = relay[b * HH + k];
  float* o = axT + ((size_t)b * 512 + feat) * LL;
  for (int l = threadIdx.x; l < LL; l += 256)
    o[l] = isrelay ? rv : data[((size_t)b * LL + l) * HH + k];
}

__global__ void transpose_fw_kernel(const float* __restrict__ Fw,
                                    float* __restrict__ FwT) {
  // Fw [NH][768][32] -> FwT [NH][32][768]
  int n = blockIdx.y;
  int idx = blockIdx.x * 256 + threadIdx.x;            // 768*32 / 256 = 96 blocks
  int k = idx >> 5, c = idx & 31;
  FwT[((size_t)n * HDD + c) * H3 + k] = Fw[((size_t)n * H3 + k) * HDD + c];
}

__global__ void ln_kernel(const float* __restrict__ nodes,
                          const float* __restrict__ g,
                          const float* __restrict__ bt,
                          float* __restrict__ xn) {
  int row  = blockIdx.x * 4 + (threadIdx.x >> 5);      // one wave per row
  int lane = threadIdx.x & 31;
  const float* x = nodes + (size_t)row * HH;
  float vals[8], s = 0.f;
  #pragma unroll
  for (int t = 0; t < 8; ++t) { vals[t] = x[lane + 32 * t]; s += vals[t]; }
  #pragma unroll
  for (int off = 16; off > 0; off >>= 1) s += __shfl_xor(s, off, 32);
  float mean = s * (1.0f / (float)HH);
  float v = 0.f;
  #pragma unroll
  for (int t = 0; t < 8; ++t) { float d = vals[t] - mean; v += d * d; }
  #pragma unroll
  for (int off = 16; off > 0; off >>= 1) v += __shfl_xor(v, off, 32);
  float rstd = rsqrtf(v * (1.0f / (float)HH) + 1e-5f);
  float* o = xn + (size_t)row * HH;
  #pragma unroll
  for (int t = 0; t < 8; ++t) {
    int c = lane + 32 * t;
    o[c] = (vals[t] - mean) * rstd * g[c] + bt[c];
  }
}

// ----------------- WMMA GEMM: hqT[n][c][bl] = (xn @ WqT + bq)T -------------
// grid (512, NH), block 128 (4 waves); wave owns 4 N-tiles, full K=256.
__global__ void gemm_hq_kernel(const float* __restrict__ xn,
                               const float* __restrict__ WQ,
                               const float* __restrict__ bq,
                               float* __restrict__ hqT) {
  int mt = blockIdx.x, n = blockIdx.y;
  int lane = threadIdx.x & 31, wv = threadIdx.x >> 5;
  int half = lane >> 4, r = lane & 15;
  const float* A  = xn + (size_t)(mt * 16 + r) * HH;
  const float* Wn = WQ + (size_t)n * HH * HH;
  v8f acc[4] = {};
  for (int kc = 0; kc < 64; ++kc) {
    int k0 = kc * 4 + half * 2;
    v2f a = *(const v2f*)(A + k0);
    #pragma unroll
    for (int nt = 0; nt < 4; ++nt) {
      int c = wv * 64 + nt * 16 + r;
      v2f bb = *(const v2f*)(Wn + (size_t)c * HH + k0);
      acc[nt] = wmma_f32(a, bb, acc[nt]);
    }
  }
  float* On = hqT + (size_t)n * HH * BLR;
  int bl0 = mt * 16 + 8 * half;
  #pragma unroll
  for (int nt = 0; nt < 4; ++nt) {
    int c = wv * 64 + nt * 16 + r;
    float bias = bq[n * HH + c];
    float* oc = On + (size_t)c * BLR + bl0;
    #pragma unroll
    for (int v = 0; v < 8; ++v) oc[v] = acc[nt][v] + bias;   // 32B-aligned run
  }
}

// ------------------------------ ei/ej dots ---------------------------------
__global__ void eiej_kernel(const float* __restrict__ hqT,
                            const float* __restrict__ axT,
                            const float* __restrict__ a1,
                            const float* __restrict__ a2,
                            float* __restrict__ ei, float* __restrict__ ej) {
  int idx = blockIdx.x * blockDim.x + threadIdx.x;     // NH*B*L threads
  int n = idx >> 13, bl = idx & 8191;
  int b = bl >> 9, l = bl & 511;
  const float* hc  = hqT + (size_t)n * HH * BLR + bl;  // stride BLR per k
  const float* axc = axT + ((size_t)b * 512) * LL + l; // stride LL per feat
  const float* A1 = a1 + n * H3;
  const float* A2 = a2 + n * H3;
  float e1 = 0.f, e2 = 0.f;
  for (int k = 0; k < HH; ++k) {
    float h = hc[(size_t)k * BLR];
    e1 += h * A1[k]; e2 += h * A2[k];
  }
  for (int k = 0; k < 2 * HH; ++k) {
    float h = axc[(size_t)k * LL];
    e1 += h * A1[HH + k]; e2 += h * A2[HH + k];
  }
  ei[idx] = e1; ej[idx] = e2;
}

// --------------------------- softmax row stats -----------------------------
__global__ void softmax_stats_kernel(const float* __restrict__ ei,
                                     const float* __restrict__ ej,
                                     const int* __restrict__ edge,
                                     float* __restrict__ mrow,
                                     float* __restrict__ isrow) {
  int row  = blockIdx.x * 4 + (threadIdx.x >> 5);      // (n*B+b)*L + i
  int lane = threadIdx.x & 31;
  int nb = row >> 9, b = nb & 15, i = row & 511;
  float eival = ei[row];
  const float* ejrow = ej + (size_t)nb * LL;
  const int*   erow  = edge + ((size_t)b * LL + i) * LL;
  float ev[16], mx = NEGV;
  #pragma unroll
  for (int t = 0; t < 16; ++t) {
    int j = lane + 32 * t;
    float e;
    if (erow[j] > 0) { float x = eival + ejrow[j]; e = x > 0.f ? x : 0.2f * x; }
    else e = NEGV;
    ev[t] = e; mx = fmaxf(mx, e);
  }
  #pragma unroll
  for (int off = 16; off > 0; off >>= 1) mx = fmaxf(mx, __shfl_xor(mx, off, 32));
  float s = 0.f;
  #pragma unroll
  for (int t = 0; t < 16; ++t) s += __expf(ev[t] - mx);
  #pragma unroll
  for (int off = 16; off > 0; off >>= 1) s += __shfl_xor(s, off, 32);
  if (lane == 0) { mrow[row] = mx; isrow[row] = 1.0f / s; }
}

// ---------------- fused att@h (WMMA) + hp@Fw + ELU epilogue ----------------
// grid (32 i-tiles, B, NH), block 128 (4 waves). Wave owns 192 cols of hp.
__global__ void gat_attn_kernel(const float* __restrict__ ei,
                                const float* __restrict__ ej,
                                const float* __restrict__ mrow,
                                const float* __restrict__ isrow,
                                const int* __restrict__ edge,
                                const float* __restrict__ hqT,
                                const float* __restrict__ axT,
                                const float* __restrict__ FwT,
                                const float* __restrict__ Fb,
                                float* __restrict__ temp) {
  __shared__ float lds_hp[16 * H3];        // 48 KB
  __shared__ float lds_red[4 * 16 * HDD];  // 8 KB
  __shared__ float lds_ej[LL];             // 2 KB
  int it = blockIdx.x, b = blockIdx.y, n = blockIdx.z;
  int tid = threadIdx.x;
  int lane = tid & 31, wv = tid >> 5;
  int half = lane >> 4, r = lane & 15;
  int nb = n * BB + b;
  size_t rowbase = (size_t)nb * LL;
  for (int t = tid; t < LL; t += 128) lds_ej[t] = ej[rowbase + t];
  __syncthreads();

  int i = it * 16 + r;
  float ei_r = ei[rowbase + i];
  float m_r  = mrow[rowbase + i];
  float is_r = isrow[rowbase + i];
  const int*   erow = edge + ((size_t)b * LL + i) * LL;
  const float* hqTn = hqT + (size_t)n * HH * BLR + (size_t)b * LL; // + c*BLR + j
  const float* axTb = axT + (size_t)b * 512 * LL;                  // + feat*LL + j
  int cw = wv * 192;
  v8f acc[12] = {};

  for (int jt = 0; jt < 32; ++jt) {
    int jb = jt * 16;
    v2f a[4];
    #pragma unroll
    for (int kk = 0; kk < 4; ++kk) {
      int j0 = jb + kk * 4 + half * 2;
      int2 ee = *(const int2*)(erow + j0);             // aligned pair
      int eflag[2] = {ee.x, ee.y};
      #pragma unroll
      for (int u = 0; u < 2; ++u) {
        int j = j0 + u;
        float e;
        if (eflag[u] > 0) { float x = ei_r + lds_ej[j]; e = x > 0.f ? x : 0.2f * x; }
        else e = NEGV;
        a[kk][u] = __expf(e - m_r) * is_r;             // softmax prob (0 if masked)
      }
    }
    #pragma unroll
    for (int nt = 0; nt < 12; ++nt) {
      int c = cw + nt * 16 + r;
      const float* colp = (c < HH) ? (hqTn + (size_t)c * BLR)
                                   : (axTb + (size_t)(c - HH) * LL);
      #pragma unroll
      for (int kk = 0; kk < 4; ++kk) {
        int j0 = jb + kk * 4 + half * 2;
        v2f bb = *(const v2f*)(colp + j0);             // single b64 load
        acc[nt] = wmma_f32(a[kk], bb, acc[nt]);
      }
    }
  }
  // hp tile -> LDS
  #pragma unroll
  for (int nt = 0; nt < 12; ++nt) {
    int c = cw + nt * 16 + r;
    #pragma unroll
    for (int v = 0; v < 8; ++v) lds_hp[(v + 8 * half) * H3 + c] = acc[nt][v];
  }
  __syncthreads();

  // out(16x32) = hp(16x768) @ Fw(768x32); wave handles its 192-K slice
  const float* FwTn = FwT + (size_t)n * HDD * H3;      // FwT[c][k]
  v8f oacc[2] = {};
  for (int kc = 0; kc < 48; ++kc) {
    int k0 = wv * 192 + kc * 4 + half * 2;
    v2f av = *(const v2f*)(lds_hp + r * H3 + k0);      // ds b64
    #pragma unroll
    for (int nt = 0; nt < 2; ++nt) {
      int c = nt * 16 + r;
      v2f bb = *(const v2f*)(FwTn + (size_t)c * H3 + k0);
      oacc[nt] = wmma_f32(av, bb, oacc[nt]);
    }
  }
  #pragma unroll
  for (int nt = 0; nt < 2; ++nt) {
    int c = nt * 16 + r;
    #pragma unroll
    for (int v = 0; v < 8; ++v)
      lds_red[(wv * 16 + v + 8 * half) * HDD + c] = oacc[nt][v];
  }
  __syncthreads();
  for (int o = tid; o < 16 * HDD; o += 128) {
    int orow = o >> 5, oc = o & 31;
    float s = lds_red[(0 * 16 + orow) * HDD + oc] + lds_red[(1 * 16 + orow) * HDD + oc]
            + lds_red[(2 * 16 + orow) * HDD + oc] + lds_red[(3 * 16 + orow) * HDD + oc];
    s += Fb[n * HDD + oc];
    s = s > 0.f ? s : (__expf(s) - 1.0f);              // ELU
    temp[((size_t)b * LL + it * 16 + orow) * (NHH * HDD) + n * HDD + oc] = s;
  }
}

// --------- nodes += leaky(temp @ fc_w + fc_b)  (in-place, WMMA) ------------
__global__ void node_update_kernel(const float* __restrict__ temp,
                                   const float* __restrict__ fcw,
                                   const float* __restrict__ fcb,
                                   float* __restrict__ nodes) {
  int mt = blockIdx.x;
  int lane = threadIdx.x & 31, wv = threadIdx.x >> 5;
  int half = lane >> 4, r = lane & 15;
  const float* A = temp + (size_t)(mt * 16 + r) * HH;
  v8f acc[4] = {};
  for (int kc = 0; kc < 64; ++kc) {
    int k0 = kc * 4 + half * 2;
    v2f a = *(const v2f*)(A + k0);
    #pragma unroll
    for (int nt = 0; nt < 4; ++nt) {
      int c = wv * 64 + nt * 16 + r;
      v2f bb; bb[0] = fcw[(size_t)k0 * HH + c]; bb[1] = fcw[(size_t)(k0 + 1) * HH + c];
      acc[nt] = wmma_f32(a, bb, acc[nt]);
    }
  }
  float* O = nodes + (size_t)(mt * 16) * HH;
  #pragma unroll
  for (int nt = 0; nt < 4; ++nt) {
    int c = wv * 64 + nt * 16 + r;
    float bias = fcb[c];
    #pragma unroll
    for (int v = 0; v < 8; ++v) {
      size_t idx = (size_t)(v + 8 * half) * HH + c;
      float t = acc[nt][v] + bias;
      t = t > 0.f ? t : 0.01f * t;
      O[idx] = O[idx] + t;
    }
  }
}

// --------------- kbuf = [relay; nodes] @ sk_wT + sk_b  (WMMA) --------------
__global__ void kproj_kernel(const float* __restrict__ relay,
                             const float* __restrict__ nodes,
                             const float* __restrict__ skw,
                             const float* __restrict__ skb,
                             float* __restrict__ kbuf) {
  int mt = blockIdx.x, b = blockIdx.y;                  // grid (33, B)
  int lane = threadIdx.x & 31, wv = threadIdx.x >> 5;
  int half = lane >> 4, r = lane & 15;
  int t = mt * 16 + r;                                  // 0..527
  bool valid = (t < 513);
  const float* Arow = (t == 0 || !valid) ? (relay + (size_t)b * HH)
                                         : (nodes + ((size_t)b * LL + (t - 1)) * HH);
  v8f acc[4] = {};
  for (int kc = 0; kc < 64; ++kc) {
    int k0 = kc * 4 + half * 2;
    v2f a;
    a[0] = valid ? Arow[k0] : 0.f;
    a[1] = valid ? Arow[k0 + 1] : 0.f;
    #pragma unroll
    for (int nt = 0; nt < 4; ++nt) {
      int c = wv * 64 + nt * 16 + r;
      v2f bb = *(const v2f*)(skw + (size_t)c * HH + k0);
      acc[nt] = wmma_f32(a, bb, acc[nt]);
    }
  }
  #pragma unroll
  for (int nt = 0; nt < 4; ++nt) {
    int c = wv * 64 + nt * 16 + r;
    float bias = skb[c];
    #pragma unroll
    for (int v = 0; v < 8; ++v) {
      int tr = mt * 16 + v + 8 * half;
      if (tr < 513) kbuf[((size_t)b * 513 + tr) * HH + c] = acc[nt][v] + bias;
    }
  }
}

// --------------------------- star attention --------------------------------
__global__ void star_att_kernel(const float* __restrict__ q,
                                const float* __restrict__ kbuf,
                                const int* __restrict__ mask,
                                float* __restrict__ attb) {
  __shared__ float lp[513];
  __shared__ float red[128];
  int b = blockIdx.x >> 3, n = blockIdx.x & 7;
  int tid = threadIdx.x;                                // block 128
  const float* qv = q + (size_t)b * HH + n * HDD;
  const float* kb = kbuf + (size_t)b * 513 * HH + n * HDD;
  const float scale = 0.17677669529663687f;             // 1/sqrt(32)
  for (int l = tid; l < 513; l += 128) {
    const float* kr = kb + (size_t)l * HH;
    float s = 0.f;
    for (int d = 0; d < HDD; ++d) s += qv[d] * kr[d];
    s *= scale;
    bool pad = (l > 0) && (mask[b * LL + l - 1] == 0);
    lp[l] = pad ? -3.0e38f : s;
  }
  __syncthreads();
  float mx = -3.0e38f;
  for (int l = tid; l < 513; l += 128) mx = fmaxf(mx, lp[l]);
  red[tid] = mx; __syncthreads();
  for (int s2 = 64; s2 > 0; s2 >>= 1) {
    if (tid < s2) red[tid] = fmaxf(red[tid], red[tid + s2]);
    __syncthreads();
  }
  mx = red[0]; __syncthreads();
  float sm = 0.f;
  for (int l = tid; l < 513; l += 128) { float p = __expf(lp[l] - mx); lp[l] = p; sm += p; }
  red[tid] = sm; __syncthreads();
  for (int s2 = 64; s2 > 0; s2 >>= 1) {
    if (tid < s2) red[tid] += red[tid + s2];
    __syncthreads();
  }
  float inv = 1.0f / red[0];
  __syncthreads();
  if (tid < HDD) {
    float acc = 0.f;
    for (int l = 0; l < 513; ++l) acc += lp[l] * kb[(size_t)l * HH + tid];
    attb[(size_t)b * HH + n * HDD + tid] = acc * inv;
  }
}

// out[b,c] = act( x[b,:] . W[c,:] + bias[c] )   (W is [256,256] "T" layout)
__global__ void vec_proj_kernel(const float* __restrict__ x,
                                const float* __restrict__ W,
                                const float* __restrict__ bias,
                                float* __restrict__ out, int leaky) {
  int b = blockIdx.x, c = threadIdx.x;                  // block 256
  const float* xr = x + (size_t)b * HH;
  const float* wr = W + (size_t)c * HH;
  float s = 0.f;
  for (int k = 0; k < HH; ++k) s += xr[k] * wr[k];
  s += bias[c];
  if (leaky) s = s > 0.f ? s : 0.01f * s;
  out[(size_t)b * HH + c] = s;
}

__global__ void mask_nodes_kernel(float* __restrict__ nodes,
                                  const int* __restrict__ mask) {
  size_t idx = (size_t)blockIdx.x * 256 + threadIdx.x;  // B*L*H elements
  int bl = (int)(idx >> 8);
  if (mask[bl] == 0) nodes[idx] = 0.0f;
}

// ------------------------------ host launch --------------------------------
extern "C" void kernel_launch(void* const* d_in, const int* in_sizes, int n_in,
                              void* d_out, int out_size, void* d_ws, size_t ws_size,
                              hipStream_t stream) {
  (void)in_sizes; (void)n_in; (void)out_size; (void)ws_size;
  const float* data   = (const float*)d_in[0];
  const float* WQw    = (const float*)d_in[1];
  const float* WQb    = (const float*)d_in[2];
  const float* a1     = (const float*)d_in[3];
  const float* a2     = (const float*)d_in[4];
  const float* Fw     = (const float*)d_in[5];
  const float* Fb     = (const float*)d_in[6];
  const float* ng     = (const float*)d_in[7];
  const float* nb_    = (const float*)d_in[8];
  const float* sqw    = (const float*)d_in[9];
  const float* sqb    = (const float*)d_in[10];
  const float* skw    = (const float*)d_in[11];
  const float* skb    = (const float*)d_in[12];
  const float* sow    = (const float*)d_in[13];
  const float* sob    = (const float*)d_in[14];
  const float* fcw    = (const float*)d_in[15];
  const float* fcb    = (const float*)d_in[16];
  const int*   maskp  = (const int*)d_in[17];
  const int*   edge   = (const int*)d_in[18];
  float* out = (float*)d_out;

  // workspace layout
  float* ws = (float*)d_ws;
  size_t off = 0;
  float* nodes = ws + off; off += (size_t)BLR * HH;
  float* xn    = ws + off; off += (size_t)BLR * HH;
  float* axT   = ws + off; off += (size_t)BLR * 2 * HH;
  float* hqT   = ws + off; off += (size_t)NHH * HH * BLR;
  float* temp  = ws + off; off += (size_t)BLR * HH;
  float* kbuf  = ws + off; off += (size_t)BB * 513 * HH;
  float* FwT   = ws + off; off += (size_t)NHH * H3 * HDD;
  float* ei    = ws + off; off += (size_t)NHH * BLR;
  float* ej    = ws + off; off += (size_t)NHH * BLR;
  float* mrow  = ws + off; off += (size_t)NHH * BLR;
  float* isrow = ws + off; off += (size_t)NHH * BLR;
  float* relayA = ws + off; off += (size_t)BB * HH;
  float* relayB = ws + off; off += (size_t)BB * HH;
  float* qbuf   = ws + off; off += (size_t)BB * HH;
  float* attb   = ws + off; off += (size_t)BB * HH;

  // nodes = data ; relay = mean(data, axis=1) ; FwT once
  hipMemcpyAsync(nodes, data, (size_t)BLR * HH * sizeof(float),
                 hipMemcpyDeviceToDevice, stream);
  relay_mean_kernel<<<BB, 256, 0, stream>>>(data, relayA);
  transpose_fw_kernel<<<dim3(96, NHH), 256, 0, stream>>>(Fw, FwT);

  float* rcur = relayA;
  float* rnext = relayB;
  for (int it = 0; it < 2; ++it) {
    const float* ng_i  = ng  + it * HH;
    const float* nb_i  = nb_ + it * HH;
    const float* sqw_i = sqw + (size_t)it * HH * HH;
    const float* sqb_i = sqb + it * HH;
    const float* skw_i = skw + (size_t)it * HH * HH;
    const float* skb_i = skb + it * HH;
    const float* sow_i = sow + (size_t)it * HH * HH;
    const float* sob_i = sob + it * HH;

    build_axT_kernel<<<BB * 512, 256, 0, stream>>>(data, rcur, axT);
    ln_kernel<<<BLR / 4, 128, 0, stream>>>(nodes, ng_i, nb_i, xn);
    gemm_hq_kernel<<<dim3(512, NHH), 128, 0, stream>>>(xn, WQw, WQb, hqT);
    eiej_kernel<<<NHH * BLR / 128, 128, 0, stream>>>(hqT, axT, a1, a2, ei, ej);
    softmax_stats_kernel<<<NHH * BLR / 4, 128, 0, stream>>>(ei, ej, edge, mrow, isrow);
    gat_attn_kernel<<<dim3(32, BB, NHH), 128, 0, stream>>>(ei, ej, mrow, isrow, edge,
                                                           hqT, axT, FwT, Fb, temp);
    node_update_kernel<<<512, 128, 0, stream>>>(temp, fcw, fcb, nodes);
    vec_proj_kernel<<<BB, 256, 0, stream>>>(rcur, sqw_i, sqb_i, qbuf, 0);
    kproj_kernel<<<dim3(33, BB), 128, 0, stream>>>(rcur, nodes, skw_i, skb_i, kbuf);
    star_att_kernel<<<BB * NHH, 128, 0, stream>>>(qbuf, kbuf, maskp, attb);
    vec_proj_kernel<<<BB, 256, 0, stream>>>(attb, sow_i, sob_i, rnext, 1);
    mask_nodes_kernel<<<BLR, 256, 0, stream>>>(nodes, maskp);

    float* tmpp = rcur; rcur = rnext; rnext = tmpp;
  }

  // outputs: nodes [B,L,H] then relay [B,H]
  hipMemcpyAsync(out, nodes, (size_t)BLR * HH * sizeof(float),
                 hipMemcpyDeviceToDevice, stream);
  hipMemcpyAsync(out + (size_t)BLR * HH, rcur, (size_t)BB * HH * sizeof(float),
                 hipMemcpyDeviceToDevice, stream);
}